// PPNPostProcessing_57346403336382
// MI455X (gfx1250) — compile-verified
//
#include <hip/hip_runtime.h>
#include <math.h>

#define TILE 256

// ---- Feature detection: gfx1250 async global<->LDS DMA builtins -----------
#if defined(__has_builtin)
#if __has_builtin(__builtin_amdgcn_global_load_async_to_lds_b128) && \
    __has_builtin(__builtin_amdgcn_global_store_async_from_lds_b128)
#define HAS_ASYNC_LDS 1
#endif
#endif
#ifndef HAS_ASYNC_LDS
#define HAS_ASYNC_LDS 0
#endif

// Builtin signature (from hipcc diagnostic): params are pointers to
// 'int __attribute__((vector_size(16)))' in AS1 (global) / AS3 (LDS).
typedef int v4i_ __attribute__((vector_size(16)));
typedef __attribute__((address_space(1))) v4i_* g_v4i_ptr;
typedef __attribute__((address_space(3))) v4i_* l_v4i_ptr;

__device__ __forceinline__ void async_g2l_b128(const void* g, void* l) {
#if HAS_ASYNC_LDS
  __builtin_amdgcn_global_load_async_to_lds_b128(
      (g_v4i_ptr)(g), (l_v4i_ptr)(l), /*offset=*/0, /*cpol=*/0);
#else
  (void)g; (void)l;
#endif
}

__device__ __forceinline__ void async_l2g_b128(void* g, const void* l) {
#if HAS_ASYNC_LDS
  __builtin_amdgcn_global_store_async_from_lds_b128(
      (g_v4i_ptr)(g), (l_v4i_ptr)(l), /*offset=*/0, /*cpol=*/0);
#else
  (void)g; (void)l;
#endif
}

__device__ __forceinline__ void wait_async0() {
#if defined(__has_builtin) && __has_builtin(__builtin_amdgcn_s_wait_asynccnt)
  __builtin_amdgcn_s_wait_asynccnt(0);
#else
  asm volatile("s_wait_asynccnt 0" ::: "memory");
#endif
}

__device__ __forceinline__ float score_pos(float l0, float l1) {
  // softmax([l0,l1])[1] == sigmoid(l1 - l0); saturates correctly at +/-inf.
  return 1.0f / (1.0f + expf(l0 - l1));
}

__global__ __launch_bounds__(TILE) void PPNPostProcessing_kernel(
    const float* __restrict__ pred,    // (n,5)
    const float* __restrict__ coords,  // (n,3)
    const int*   __restrict__ batch,   // (n,)
    float* __restrict__ out,           // (n,5)
    float* __restrict__ keep_out,      // (n,)
    int n) {
  const int t     = threadIdx.x;
  const int tile0 = blockIdx.x * TILE;

#if HAS_ASYNC_LDS
  // Fast path: full tile, keep-region base 16B-aligned (n % 4 == 0).
  if (tile0 + TILE <= n && (n & 3) == 0) {
    __shared__ float4 s_pred4[TILE * 5 / 4];   // 5120 B
    __shared__ float4 s_coord4[TILE * 3 / 4];  // 3072 B
    __shared__ int4   s_batch4[TILE / 4];      // 1024 B
    __shared__ float4 s_out4[TILE * 5 / 4];    // 5120 B
    __shared__ float4 s_keep4[TILE / 4];       // 1024 B
    float* s_pred  = (float*)s_pred4;
    float* s_coord = (float*)s_coord4;
    int*   s_batch = (int*)s_batch4;
    float* s_out   = (float*)s_out4;
    float* s_keep  = (float*)s_keep4;

    // -------- Stage in: contiguous b128 async DMA (global -> LDS) --------
    const float4* gp = (const float4*)(pred   + (size_t)tile0 * 5); // 320 f4
    const float4* gc = (const float4*)(coords + (size_t)tile0 * 3); // 192 f4
    const int4*   gb = (const int4*)(batch + tile0);                //  64 i4

    async_g2l_b128(gp + t, s_pred4 + t);                          // f4 0..255
    if (t < 64)  async_g2l_b128(gp + 256 + t, s_pred4 + 256 + t); // f4 256..319
    if (t < 192) async_g2l_b128(gc + t, s_coord4 + t);
    if (t < 64)  async_g2l_b128(gb + t, s_batch4 + t);

    wait_async0();
    __syncthreads();

    // -------- Compute (LDS strides 5 and 3 are bank-conflict free) --------
    const float p0 = s_pred[t * 5 + 0];
    const float p1 = s_pred[t * 5 + 1];
    const float p2 = s_pred[t * 5 + 2];
    const float l0 = s_pred[t * 5 + 3];
    const float l1 = s_pred[t * 5 + 4];
    const float c0 = s_coord[t * 3 + 0];
    const float c1 = s_coord[t * 3 + 1];
    const float c2 = s_coord[t * 3 + 2];
    const float bf = (float)s_batch[t];

    const float sc   = score_pos(l0, l1);
    const bool  keep = sc > 0.9f;

    s_out[t * 5 + 0] = keep ? (c0 + 0.5f + p0) : 0.0f;
    s_out[t * 5 + 1] = keep ? (c1 + 0.5f + p1) : 0.0f;
    s_out[t * 5 + 2] = keep ? (c2 + 0.5f + p2) : 0.0f;
    s_out[t * 5 + 3] = keep ? bf : 0.0f;
    s_out[t * 5 + 4] = keep ? sc : 0.0f;
    s_keep[t]        = keep ? 1.0f : 0.0f;

    __syncthreads();

    // -------- Stage out: contiguous b128 async DMA (LDS -> global) --------
    float4* go = (float4*)(out + (size_t)tile0 * 5);   // 320 f4
    float4* gk = (float4*)(keep_out + tile0);          //  64 f4
    async_l2g_b128(go + t, s_out4 + t);
    if (t < 64) async_l2g_b128(go + 256 + t, s_out4 + 256 + t);
    if (t < 64) async_l2g_b128(gk + t, s_keep4 + t);
    // s_endpgm performs an implicit wait-idle, draining ASYNCcnt.
    return;
  }
#endif

  // -------- Fallback / tail path: direct streaming with NT hints --------
  const int i = tile0 + t;
  if (i < n) {
    const float* p = pred + (size_t)i * 5;
    const float* c = coords + (size_t)i * 3;
    const float p0 = p[0], p1 = p[1], p2 = p[2], l0 = p[3], l1 = p[4];
    const float c0 = c[0], c1 = c[1], c2 = c[2];
    const float bf = (float)batch[i];

    const float sc   = score_pos(l0, l1);
    const bool  keep = sc > 0.9f;

    float* o = out + (size_t)i * 5;
    __builtin_nontemporal_store(keep ? (c0 + 0.5f + p0) : 0.0f, o + 0);
    __builtin_nontemporal_store(keep ? (c1 + 0.5f + p1) : 0.0f, o + 1);
    __builtin_nontemporal_store(keep ? (c2 + 0.5f + p2) : 0.0f, o + 2);
    __builtin_nontemporal_store(keep ? bf : 0.0f, o + 3);
    __builtin_nontemporal_store(keep ? sc : 0.0f, o + 4);
    __builtin_nontemporal_store(keep ? 1.0f : 0.0f, keep_out + i);
  }
}

extern "C" void kernel_launch(void* const* d_in, const int* in_sizes, int n_in,
                              void* d_out, int out_size, void* d_ws, size_t ws_size,
                              hipStream_t stream) {
  const float* pred   = (const float*)d_in[0];
  const float* coords = (const float*)d_in[1];
  const int*   batch  = (const int*)d_in[2];
  const int    n      = in_sizes[2];  // batch_ids element count == N

  float* out      = (float*)d_out;            // (n,5) flattened
  float* keep_out = out + (size_t)n * 5;      // (n,) keep flags, concatenated

  const int blocks = (n + TILE - 1) / TILE;
  hipLaunchKernelGGL(PPNPostProcessing_kernel, dim3(blocks), dim3(TILE), 0,
                     stream, pred, coords, batch, out, keep_out, n);
}